// CircuitLayer_57183194579635
// MI455X (gfx1250) — compile-verified
//
#include <hip/hip_runtime.h>
#include <math.h>

// ---------------------------------------------------------------------------
// Segmented log-sum-exp over sorted segment ids (forward of CircuitLayer).
//   out[s] = log( sum_{e: ix_out[e]==s} exp(x[e] - mx[s]) + eps ) + mx[s]
// ix_out is sorted -> each segment's edges are a contiguous run [start, end).
// Memory-bound: ~256MB mandatory reads -> ~11us floor at 23.3 TB/s.
// Deterministic: no atomics; boundary writes unique; fixed reduce order.
// gfx1250 paths: global_load_async_to_lds_b128 th:TH_LOAD_NT (+
// s_wait_asynccnt), global_prefetch_b8 (x=128MB is kept resident in the
// 192MB L2 while the single-use ix stream bypasses with NT hints).
// ---------------------------------------------------------------------------

static constexpr float kEps = 1e-31f;
static constexpr unsigned kNoSeg = 0xFFFFFFFFu;
static constexpr int kTile = 2048;  // edges per block in k_bounds (256 thr * 8)

static __device__ __forceinline__ float wave_max32(float v) {
#pragma unroll
  for (int off = 16; off; off >>= 1)
    v = fmaxf(v, __shfl_xor(v, off, 32));
  return v;
}

static __device__ __forceinline__ float wave_sum32(float v) {
#pragma unroll
  for (int off = 16; off; off >>= 1)
    v += __shfl_xor(v, off, 32);
  return v;
}

// Kernel 0: sentinel-init segment start table (must re-init every call for
// graph-replay determinism; ws is poisoned once and never restored).
__global__ void k_init(unsigned* __restrict__ seg_start, int S) {
  int s = blockIdx.x * blockDim.x + threadIdx.x;
  if (s < S) seg_start[s] = kNoSeg;
}

// Kernel 1: stream ix_out tile-by-tile through LDS via the CDNA5 async
// global->LDS copy engine (non-temporal: ix is single-use and must not evict
// the prefetched x stream from L2), then emit segment [start,end) bounds with
// plain unique-writer stores. Halo cells live at the end of the LDS tile so
// the b128 async writes stay 16B-aligned.
__global__ void __launch_bounds__(256) k_bounds(
    const int* __restrict__ ix, const float* __restrict__ x,
    unsigned* __restrict__ seg_start, unsigned* __restrict__ seg_end,
    long long E) {
  __shared__ __align__(16) int lds_ix[kTile + 2];
  const int t = threadIdx.x;
  const long long tile = (long long)blockIdx.x * kTile;
  const long long base = tile + (long long)t * 8;

  if (tile + kTile <= E) {
    // ---- fast path: whole tile in range ----
    // gfx1250: global_prefetch_b8 -- warm L2 with the x stream for k_reduce.
    __builtin_prefetch(x + base, 0, 3);

    // Async-stage 8 ints/lane into LDS (2 x b128, NT). INST_OFFSET is added
    // to both the LDS and global addresses, so one address pair covers both.
    unsigned lds_off = (unsigned)(uintptr_t)(&lds_ix[t * 8]);
    unsigned long long gaddr = (unsigned long long)(uintptr_t)(ix + base);
    asm volatile(
        "global_load_async_to_lds_b128 %0, %1, off th:TH_LOAD_NT\n\t"
        "global_load_async_to_lds_b128 %0, %1, off offset:16 th:TH_LOAD_NT"
        :
        : "v"(lds_off), "v"(gaddr)
        : "memory");

    // Halo: previous edge of the tile and first edge of the next tile.
    if (t == 0)
      lds_ix[kTile] = (tile == 0) ? -1 : __builtin_nontemporal_load(ix + tile - 1);
    if (t == 1)
      lds_ix[kTile + 1] =
          (tile + kTile >= E) ? -2 : __builtin_nontemporal_load(ix + tile + kTile);

    asm volatile("s_wait_asynccnt 0x0" ::: "memory");
    __syncthreads();

    int v[10];
    v[0] = (t == 0) ? lds_ix[kTile] : lds_ix[t * 8 - 1];
    const int4* l4 = reinterpret_cast<const int4*>(&lds_ix[t * 8]);
    int4 a = l4[0];
    int4 b = l4[1];
    v[1] = a.x; v[2] = a.y; v[3] = a.z; v[4] = a.w;
    v[5] = b.x; v[6] = b.y; v[7] = b.z; v[8] = b.w;
    v[9] = (t == 255) ? lds_ix[kTile + 1] : lds_ix[t * 8 + 8];

#pragma unroll
    for (int j = 1; j <= 8; ++j) {
      long long e = base + (j - 1);
      if (v[j] != v[j - 1]) seg_start[v[j]] = (unsigned)e;
      if (v[j] != v[j + 1]) seg_end[v[j]] = (unsigned)(e + 1);
    }
  } else {
    // ---- tail path (E not a tile multiple): scalar, same semantics ----
    for (long long e = base; e < E && e < base + 8; ++e) {
      int c = __builtin_nontemporal_load(ix + e);
      int pv = (e == 0) ? -1 : __builtin_nontemporal_load(ix + e - 1);
      int nv = (e == E - 1) ? -2 : __builtin_nontemporal_load(ix + e + 1);
      if (c != pv) seg_start[c] = (unsigned)e;
      if (c != nv) seg_end[c] = (unsigned)(e + 1);
    }
  }
}

// Kernel 2: one wave32 per segment (mean segment length == 32 edges here --
// a perfect fit for CDNA5 wave32). Two lane-strided sweeps over the
// segment's contiguous x range (L2-hot from the prefetch; second sweep is
// L0-hot): max, then sum(exp(x-max)). Fixed cross-lane combine order ->
// bitwise deterministic across graph replays.
__global__ void k_reduce(const float* __restrict__ x,
                         const unsigned* __restrict__ seg_start,
                         const unsigned* __restrict__ seg_end,
                         float* __restrict__ out, int S) {
  int wave = (int)((blockIdx.x * blockDim.x + threadIdx.x) >> 5);
  int lane = threadIdx.x & 31;
  if (wave >= S) return;

  unsigned lo = seg_start[wave];
  if (lo == kNoSeg) {  // empty segment: segment_max identity is -inf
    if (lane == 0) __builtin_nontemporal_store(-__builtin_inff(), out + wave);
    return;
  }
  unsigned hi = seg_end[wave];

  float m = -__builtin_inff();
  for (unsigned e = lo + lane; e < hi; e += 32) m = fmaxf(m, x[e]);
  m = wave_max32(m);

  float s = 0.0f;
  for (unsigned e = lo + lane; e < hi; e += 32) s += __expf(x[e] - m);
  s = wave_sum32(s);

  if (lane == 0)
    __builtin_nontemporal_store(__logf(s + kEps) + m, out + wave);
}

extern "C" void kernel_launch(void* const* d_in, const int* in_sizes, int n_in,
                              void* d_out, int out_size, void* d_ws, size_t ws_size,
                              hipStream_t stream) {
  const float* x = (const float*)d_in[0];
  const int* ix_out = (const int*)d_in[1];
  // d_in[2] (ix_in) is unused in the forward pass.
  float* out = (float*)d_out;

  const long long E = (long long)in_sizes[0];
  const int S = out_size;

  unsigned* seg_start = (unsigned*)d_ws;  // S u32
  unsigned* seg_end = seg_start + S;      // S u32  (8MB total @ S=1M)

  const int tb = 256;

  k_init<<<(unsigned)((S + tb - 1) / tb), tb, 0, stream>>>(seg_start, S);

  unsigned bblocks = (unsigned)((E + kTile - 1) / kTile);
  k_bounds<<<bblocks, tb, 0, stream>>>(ix_out, x, seg_start, seg_end, E);

  long long rthreads = (long long)S * 32;
  k_reduce<<<(unsigned)((rthreads + tb - 1) / tb), tb, 0, stream>>>(
      x, seg_start, seg_end, out, S);
}